// NsoltBlockIdct2dLayer_49727131353834
// MI455X (gfx1250) — compile-verified
//
#include <hip/hip_runtime.h>

typedef float v2f __attribute__((ext_vector_type(2)));
typedef float v4f __attribute__((ext_vector_type(4)));
typedef float v8f __attribute__((ext_vector_type(8)));

#define NW 8              // waves per workgroup
#define TPW 16            // tiles per wave
#define TILES_PER_CH 32768 // 32 samples * 128 rows * (128/16) col-tiles
#define TSTRIDE 68         // padded LDS stride (floats) per block column

// Orthonormal 8-point DCT-II matrix entry C[k][m] = 0.5*cos(pi*(m+0.5)*k/8), row0 * 1/sqrt(2)
__device__ __forceinline__ float dctC(int k, int m) {
  if (k == 0) return 0.35355339059327379f;            // 0.5 / sqrt(2)
  return 0.5f * cosf(0.19634954084936207f * (float)((2 * m + 1) * k)); // pi/16*(2m+1)*k
}

__global__ __launch_bounds__(256) void nsolt_idct_kernel(
    const float* __restrict__ in0, const float* __restrict__ in1,
    const float* __restrict__ in2, float* __restrict__ out) {
  __shared__ float sG[4096];              // 64x64 composite perm+IDCT operator, row-major
  __shared__ float sT[NW * 16 * TSTRIDE]; // per-wave output transpose staging

  // Build G: y[p=8m+n] = sum_q G[p][q] * coef[q]; (k,l) from quadrant packing.
  for (int idx = threadIdx.x; idx < 4096; idx += 256) {
    int p = idx >> 6, q = idx & 63;
    int m = p >> 3, n = p & 7;
    int grp = q >> 4, a = (q >> 2) & 3, b = q & 3;
    int k = 2 * a + ((grp == 1 || grp == 2) ? 1 : 0);  // ee|oo|oe|eo quadrants
    int l = 2 * b + ((grp == 1 || grp == 3) ? 1 : 0);
    sG[idx] = dctC(k, m) * dctC(l, n);
  }
  __syncthreads();

  const int lane = threadIdx.x & 31;
  const int half = lane >> 4;   // 0: lanes 0-15, 1: lanes 16-31
  const int lrow = lane & 15;
  const int wave = threadIdx.x >> 5;

  // Hoist A operand (G slices) into registers in WMMA 16x4 f32 A-layout:
  // VGPR0 = A[M=lrow][K = 0|2 (half)], VGPR1 = A[lrow][K = 1|3]  -> v2f per (mc,k)
  v2f A[4][16];
#pragma unroll
  for (int mc = 0; mc < 4; ++mc) {
#pragma unroll
    for (int k = 0; k < 16; ++k) {
      A[mc][k] = *(const v2f*)&sG[(16 * mc + lrow) * 64 + 4 * k + 2 * half];
    }
  }

  float* myT = &sT[wave * 16 * TSTRIDE];
  const int tbase = ((int)blockIdx.x * NW + wave) * TPW;

  for (int it = 0; it < TPW; ++it) {
    int tile = tbase + it;
    int ch = tile / TILES_PER_CH;
    int rem = tile - ch * TILES_PER_CH;
    int s  = rem >> 10;          // / (128*8)
    int r  = (rem >> 3) & 127;
    int ct = rem & 7;            // 16-block column tile

    const float* src = (ch == 0) ? in0 : ((ch == 1) ? in1 : in2);
    // B operand: lane n holds block (ct*16+lrow); K split across wave halves.
    const float* bptr = src + (((s * 128 + r) * 128 + ct * 16 + lrow) << 6) + half * 2;

    v8f acc[4] = {};
#pragma unroll
    for (int k = 0; k < 16; ++k) {
      v2f b = *(const v2f*)(bptr + 4 * k);   // coefs q = 4k+2*half, +1
#pragma unroll
      for (int mc = 0; mc < 4; ++mc) {
        acc[mc] = __builtin_amdgcn_wmma_f32_16x16x4_f32(
            false, A[mc][k], false, b, (short)0, acc[mc], false, false);
      }
    }

    // D layout: VGPR g, lanes0-15 -> p=16mc+g, blk=lane; lanes16-31 -> p=16mc+g+8.
    // Scatter into LDS as lds[blk*68 + p] (16B-aligned b128 stores).
    float* tb = myT + lrow * TSTRIDE + half * 8;
#pragma unroll
    for (int mc = 0; mc < 4; ++mc) {
      *(v4f*)(tb + 16 * mc)     = acc[mc].lo;
      *(v4f*)(tb + 16 * mc + 4) = acc[mc].hi;
    }
    asm volatile("s_wait_dscnt 0x0" ::: "memory");  // same-wave LDS RAW fence

    // Coalesced output: 8 rows x 512B (16 blocks * 8 cols) per tile.
    const float* lsrc = myT + (lane >> 1) * TSTRIDE + (lane & 1) * 4;
    float* op = out + (((((s * 3 + ch) << 10) + r * 8) << 10) + ct * 128 + lane * 4);
#pragma unroll
    for (int m = 0; m < 8; ++m) {
      *(v4f*)(op + (m << 10)) = *(const v4f*)(lsrc + 8 * m);
    }
    asm volatile("s_wait_dscnt 0x0" ::: "memory");  // WAR fence before next iter reuses myT
  }
}

extern "C" void kernel_launch(void* const* d_in, const int* in_sizes, int n_in,
                              void* d_out, int out_size, void* d_ws, size_t ws_size,
                              hipStream_t stream) {
  const float* in0 = (const float*)d_in[0];
  const float* in1 = (const float*)d_in[1];
  const float* in2 = (const float*)d_in[2];
  float* out = (float*)d_out;
  dim3 grid(3 * TILES_PER_CH / (NW * TPW));  // 768 workgroups
  dim3 block(NW * 32);                       // 256 threads = 8 waves
  nsolt_idct_kernel<<<grid, block, 0, stream>>>(in0, in1, in2, out);
}